// GraphConvolution_11579231830756
// MI455X (gfx1250) — compile-verified
//
#include <hip/hip_runtime.h>

#define N_NODES 50000
#define NNZ_E   800000
#define NSUP    3
#define D_IN    512
#define D_OUT   256

typedef __attribute__((ext_vector_type(2))) float v2f;
typedef __attribute__((ext_vector_type(8))) float v8f;
typedef unsigned int u32x4 __attribute__((ext_vector_type(4)));
typedef int          i32x8 __attribute__((ext_vector_type(8)));
typedef int          i32x4 __attribute__((ext_vector_type(4)));

#if defined(__has_builtin)
#if __has_builtin(__builtin_amdgcn_tensor_load_to_lds)
#define USE_TDM 1
#endif
#endif
#ifndef USE_TDM
#define USE_TDM 0
#endif

#if USE_TDM
// ---------------------------------------------------------------------------
// Issue one 2D TDM tile load (D# per CDNA5 ISA).
// dims/strides in 4-byte elements; pad_interval/amount encode LDS row padding.
// 6-arg builtin form (clang-23 / therock-10.0): (g0, g1, g2, g3, g4, cpol).
// ---------------------------------------------------------------------------
static __device__ __forceinline__ void tdm_load_2d(
    unsigned lds_addr, const void* gaddr,
    unsigned tile_d0, unsigned tile_d1,
    unsigned tensor_d0, unsigned tensor_d1,
    unsigned long long stride0,
    unsigned pad_en, unsigned pad_ivl, unsigned pad_amt) {
  const unsigned long long ga = (unsigned long long)(uintptr_t)gaddr;
  u32x4 g0;
  g0.x = 1u;                                         // count=1 (valid user D#)
  g0.y = lds_addr;                                   // LDS byte address
  g0.z = (unsigned)(ga & 0xFFFFFFFFu);               // global_addr[31:0]
  g0.w = (unsigned)((ga >> 32) & 0x1FFFFFFu) | (2u << 30);  // addr[56:32] | type=2
  i32x8 g1;
  g1[0] = (int)((2u << 16) | (pad_en << 20) | (pad_ivl << 22) | (pad_amt << 25)); // data_size=4B
  g1[1] = (int)((tensor_d0 & 0xFFFFu) << 16);        // tensor_dim0[15:0]
  g1[2] = (int)(((tensor_d0 >> 16) & 0xFFFFu) | ((tensor_d1 & 0xFFFFu) << 16));
  g1[3] = (int)(((tensor_d1 >> 16) & 0xFFFFu) | ((tile_d0 & 0xFFFFu) << 16));
  g1[4] = (int)(tile_d1 & 0xFFFFu);                  // tile_dim1 | tile_dim2=0
  g1[5] = (int)(unsigned)(stride0 & 0xFFFFFFFFull);  // tensor_dim0_stride[31:0]
  g1[6] = (int)(unsigned)((stride0 >> 32) & 0xFFFFull);
  g1[7] = 0;
  const i32x4 gz4 = {0, 0, 0, 0};                    // 2D: groups 2/3 unused
  const i32x8 gz8 = {0, 0, 0, 0, 0, 0, 0, 0};
  __builtin_amdgcn_tensor_load_to_lds(g0, g1, gz4, gz4, gz8, 0);
}
#endif

// ---------------------------------------------------------------------------
// Kernel 1: pre[s] = x @ W[s]   (fp32 WMMA 16x16x4, TDM-fed double buffering)
// Block tile: M=64, N=128, Kc=32. 8 waves: 4(M) x 2(N), 16x64 strip per wave.
// ---------------------------------------------------------------------------
__global__ __launch_bounds__(256) void gcn_gemm_wmma(
    const float* __restrict__ x, const float* __restrict__ W,
    float* __restrict__ pre) {
  __shared__ float xT[2][64][33];   // row pad via TDM pad (32 dw + 1 dw)
  __shared__ float wT[2][32][128];

  const int s    = blockIdx.z;
  const int m0   = blockIdx.x * 64;
  const int n0   = blockIdx.y * 128;
  const int t    = threadIdx.x;
  const int lane = t & 31;
  const int wave = t >> 5;
  const int waveM = wave & 3;
  const int waveN = wave >> 2;
  const int l    = lane & 15;
  const int half = lane >> 4;

  v8f acc[4] = {};
  const float* Ws = W + (size_t)s * D_IN * D_OUT;
  const int rowsLeft = N_NODES - m0;
  const unsigned tdimM = (unsigned)(rowsLeft < 64 ? rowsLeft : 64);

#if USE_TDM
  // Stage K-chunk kc into buffer kc&1 with two tensor DMAs (issued by wave 0;
  // TDM ignores EXEC, branch is wave-uniform).
#define STAGE(kc)                                                              \
  do {                                                                         \
    const int _k0 = (kc) * 32, _b = (kc) & 1;                                  \
    tdm_load_2d((unsigned)(uintptr_t)&xT[_b][0][0],                            \
                x + (size_t)m0 * D_IN + _k0, 32u, 64u, 32u, tdimM,             \
                (unsigned long long)D_IN, 1u, 4u, 0u);  /* pad 32dw -> +1dw */ \
    tdm_load_2d((unsigned)(uintptr_t)&wT[_b][0][0],                            \
                Ws + (size_t)_k0 * D_OUT + n0, 128u, 32u, 128u, 32u,           \
                (unsigned long long)D_OUT, 0u, 0u, 0u);                        \
  } while (0)

  if (wave == 0) STAGE(0);
  for (int kc = 0; kc < D_IN / 32; ++kc) {
    if (wave == 0) __builtin_amdgcn_s_wait_tensorcnt(0);  // s_wait_tensorcnt
    __syncthreads();                                       // chunk kc visible
    if (wave == 0 && kc + 1 < D_IN / 32) STAGE(kc + 1);    // overlap next DMA
    const int b = kc & 1;
#else
  for (int kc = 0; kc < D_IN / 32; ++kc) {
    const int b = kc & 1;
    const int k0 = kc * 32;
#pragma unroll
    for (int i = 0; i < 2; ++i) {                  // x tile 64x32
      int idx = t * 2 + i, r = idx >> 3, c4 = (idx & 7) * 4, gr = m0 + r;
      float4 v = make_float4(0.f, 0.f, 0.f, 0.f);
      if (gr < N_NODES) v = *(const float4*)(x + (size_t)gr * D_IN + k0 + c4);
      xT[b][r][c4 + 0] = v.x; xT[b][r][c4 + 1] = v.y;
      xT[b][r][c4 + 2] = v.z; xT[b][r][c4 + 3] = v.w;
    }
#pragma unroll
    for (int i = 0; i < 4; ++i) {                  // W tile 32x128
      int idx = t * 4 + i, r = idx >> 5, c4 = (idx & 31) * 4;
      *(float4*)&wT[b][r][c4] =
          *(const float4*)(Ws + (size_t)(k0 + r) * D_OUT + n0 + c4);
    }
    __syncthreads();
#endif

#pragma unroll
    for (int kk = 0; kk < 8; ++kk) {               // 8 x (K=4)
      const int kb = kk * 4 + half * 2;
      v2f a;                                        // A 16x4 lane striping
      a.x = xT[b][waveM * 16 + l][kb];
      a.y = xT[b][waveM * 16 + l][kb + 1];
#pragma unroll
      for (int nt = 0; nt < 4; ++nt) {
        const int col = waveN * 64 + nt * 16 + l;
        v2f bb;
        bb.x = wT[b][kb][col];
        bb.y = wT[b][kb + 1][col];
        acc[nt] = __builtin_amdgcn_wmma_f32_16x16x4_f32(
            false, a, false, bb, (short)0, acc[nt], false, false);
      }
    }
    __syncthreads();                               // buffer safe for reuse
  }

  float* preS = pre + (size_t)s * N_NODES * D_OUT;
#pragma unroll
  for (int nt = 0; nt < 4; ++nt) {
    const int col = n0 + waveN * 64 + nt * 16 + l;
#pragma unroll
    for (int v = 0; v < 8; ++v) {
      const int row = m0 + waveM * 16 + v + half * 8;
      if (row < N_NODES) preS[(size_t)row * D_OUT + col] = acc[nt][v];
    }
  }
}

// ---------------------------------------------------------------------------
__global__ void gcn_zero(float* __restrict__ out, int n) {
  int i = blockIdx.x * blockDim.x + threadIdx.x;
  if (i < n) out[i] = 0.f;
}
__global__ void gcn_relu(float* __restrict__ out, int n) {
  int i = blockIdx.x * blockDim.x + threadIdx.x;
  if (i < n) out[i] = fmaxf(out[i], 0.f);
}

// ---------------------------------------------------------------------------
// Kernel 2: COO scatter  out[r] += cw[s]*vals[e] * pre[s][c]   (L2-bound)
// ---------------------------------------------------------------------------
__global__ __launch_bounds__(256) void gcn_spmm(
    const int* __restrict__ rows, const int* __restrict__ cols,
    const float* __restrict__ vals, const float* __restrict__ cw,
    const float* __restrict__ pre, float* __restrict__ out,
    int edgesPerBlock) {
  const int s  = blockIdx.y;
  const int t  = threadIdx.x;
  const float c = cw[s];
  int e0 = blockIdx.x * edgesPerBlock;
  int e1 = e0 + edgesPerBlock;
  if (e1 > NNZ_E) e1 = NNZ_E;

  const int*   r    = rows + (size_t)s * NNZ_E;
  const int*   cc   = cols + (size_t)s * NNZ_E;
  const float* v    = vals + (size_t)s * NNZ_E;
  const float* preS = pre  + (size_t)s * N_NODES * D_OUT;

  for (int e = e0; e < e1; ++e) {
    const int   ri = r[e];
    const int   ci = cc[e];
    const float w  = v[e] * c;
    if (e + 1 < e1)
      __builtin_prefetch(preS + (size_t)cc[e + 1] * D_OUT + t, 0, 0);
    atomicAdd(out + (size_t)ri * D_OUT + t, w * preS[(size_t)ci * D_OUT + t]);
  }
}

// ---------------------------------------------------------------------------
extern "C" void kernel_launch(void* const* d_in, const int* in_sizes, int n_in,
                              void* d_out, int out_size, void* d_ws, size_t ws_size,
                              hipStream_t stream) {
  const float* x    = (const float*)d_in[0];
  const float* W    = (const float*)d_in[1];
  const int*   rows = (const int*)  d_in[2];
  const int*   cols = (const int*)  d_in[3];
  const float* vals = (const float*)d_in[4];
  const float* cw   = (const float*)d_in[5];
  float*       out  = (float*)d_out;
  float*       pre  = (float*)d_ws;   // [3, N, 256] scratch (153.6 MB)

  const int nOut = N_NODES * D_OUT;
  gcn_zero<<<(nOut + 255) / 256, 256, 0, stream>>>(out, nOut);

  dim3 gGrid((N_NODES + 63) / 64, D_OUT / 128, NSUP);
  gcn_gemm_wmma<<<gGrid, 256, 0, stream>>>(x, W, pre);

  const int epb = 512;
  dim3 sGrid((NNZ_E + epb - 1) / epb, NSUP);
  gcn_spmm<<<sGrid, 256, 0, stream>>>(rows, cols, vals, cw, pre, out, epb);

  gcn_relu<<<(nOut + 255) / 256, 256, 0, stream>>>(out, nOut);
}